// String_Finder_53790170415242
// MI455X (gfx1250) — compile-verified
//
#include <hip/hip_runtime.h>
#include <math.h>

typedef __attribute__((ext_vector_type(2))) float v2f;
typedef __attribute__((ext_vector_type(8))) float v8f;

#define NIMG 16
#define HH   512
#define WW   512
#define HWN  (HH*WW)          // 262144 = 2^18
#define NHW  (NIMG*HWN)       // 4194304

__device__ __forceinline__ int reflect_idx(int i, int n) {
  if (i < 0)  i = -i;
  if (i >= n) i = 2*n - 2 - i;
  return i;
}

// ---------------- scalar init ----------------
__global__ void kce_init(unsigned* sc) {
  if (threadIdx.x == 0) { sc[0] = 0xFFFFFFFFu; sc[1] = 0u; }
}

// ---------------- global min over images ----------------
__global__ __launch_bounds__(256) void kce_min(const float* __restrict__ img, unsigned* sc, int n) {
  __shared__ unsigned red[256];
  unsigned best = 0xFFFFFFFFu;
  for (int i = blockIdx.x * blockDim.x + threadIdx.x; i < n; i += gridDim.x * blockDim.x) {
    unsigned b = __float_as_uint(img[i]);
    unsigned key = (b & 0x80000000u) ? ~b : (b | 0x80000000u);  // order-preserving map
    best = best < key ? best : key;
  }
  red[threadIdx.x] = best; __syncthreads();
  for (int s = 128; s > 0; s >>= 1) {
    if (threadIdx.x < s) { unsigned o = red[threadIdx.x + s]; if (o < red[threadIdx.x]) red[threadIdx.x] = o; }
    __syncthreads();
  }
  if (threadIdx.x == 0) atomicMin(&sc[0], red[0]);
}

// ---------------- mag = ||rgb||, global max ----------------
__global__ __launch_bounds__(256) void kce_mag(const float* __restrict__ img, unsigned* sc,
                                               float* __restrict__ mag) {
  unsigned kmin = sc[0];
  float fmin = (kmin & 0x80000000u) ? __uint_as_float(kmin & 0x7FFFFFFFu)
                                    : __uint_as_float(~kmin);
  const bool remap = fmin < -0.001f;

  __shared__ unsigned red[256];
  int i = blockIdx.x * 256 + threadIdx.x;          // exact cover of NHW
  int n = i >> 18;
  int p = i & (HWN - 1);
  const float* im = img + (size_t)n * 3 * HWN;
  float r = im[p], g = im[HWN + p], b = im[2*HWN + p];
  if (remap) { r = (r + 1.0f) * 0.5f; g = (g + 1.0f) * 0.5f; b = (b + 1.0f) * 0.5f; }
  float m = sqrtf(r*r + g*g + b*b);
  mag[i] = m;
  red[threadIdx.x] = __float_as_uint(m);           // non-negative: bits compare as uint
  __syncthreads();
  for (int s = 128; s > 0; s >>= 1) {
    if (threadIdx.x < s) { unsigned o = red[threadIdx.x + s]; if (o > red[threadIdx.x]) red[threadIdx.x] = o; }
    __syncthreads();
  }
  if (threadIdx.x == 0) atomicMax(&sc[1], red[0]);
}

// ---------------- sobel via V_WMMA_F32_16X16X4_F32 ----------------
// Each wave computes one 16x16 output tile. conv5x5 = sum_dy [ A(16x20 shifted rows) x B_banded(20x16) ]
__global__ __launch_bounds__(256) void kce_sobel(const float* __restrict__ mag,
                                                 const unsigned* __restrict__ sc,
                                                 float* __restrict__ out,
                                                 float* __restrict__ grad,
                                                 unsigned char* __restrict__ phase) {
  __shared__ float tile[8][20 * 21];   // per-wave 20x20 staging (stride 21)
  __shared__ float tw[2][5][36];       // banded weight tables: [0]=S (sobel_y), [1]=S^T (sobel_x)

  const int t    = threadIdx.x;
  const int wave = t >> 5;
  const int lane = t & 31;
  const int lhi  = lane >> 4;          // 0/1
  const int lm   = lane & 15;

  // zero weight tables
  for (int i = t; i < 2 * 5 * 36; i += 256) ((float*)tw)[i] = 0.0f;
  __syncthreads();
  // fill S[a][b] = (a-2) / den / 6, den = (b==2) ? 1 : (a-2)^2+(b-2)^2
  if (t < 25) {
    int a = t / 5, b = t % 5;
    float ra = (float)(a - 2), rb = (float)(b - 2);
    float den = (b == 2) ? 1.0f : (ra * ra + rb * rb);
    float wv = ra / den / 6.0f;
    tw[0][a][16 + b] = wv;             // W_SY[dy=a][dx=b]
    tw[1][b][16 + a] = wv;             // W_SX[dy][dx] = S[dx][dy]
  }

  const int tileId = blockIdx.x * 8 + wave;        // 16384 tiles total
  const int n   = tileId >> 10;
  const int rem = tileId & 1023;
  const int r0  = (rem >> 5) << 4;
  const int c0  = (rem & 31) << 4;
  const float maxv = __uint_as_float(sc[1]);
  const int base = n * HWN;

  float* T = tile[wave];
  for (int i = lane; i < 400; i += 32) {
    int row = i / 20, col = i % 20;
    int gy = reflect_idx(r0 + row - 2, HH);
    int gx = reflect_idx(c0 + col - 2, WW);
    T[row * 21 + col] = mag[base + gy * WW + gx] / maxv;   // normalized, reflect-padded
  }
  __syncthreads();

  v8f accY = {}; v8f accX = {};
  #pragma unroll
  for (int dy = 0; dy < 5; ++dy) {
    const float* trow = &T[(lm + dy) * 21];
    #pragma unroll
    for (int ch = 0; ch < 5; ++ch) {
      // A fragment: A[r=lm][k], k = v + 2*lhi, tile col = ch*4 + k
      v2f a;
      a.x = trow[ch * 4 + 2 * lhi];
      a.y = trow[ch * 4 + 2 * lhi + 1];
      // B fragment: B[k][c=lm] = w[dy][j - c], j = ch*4 + k  (banded, zero elsewhere)
      int j0 = ch * 4 + 2 * lhi;
      v2f bY, bX;
      bY.x = tw[0][dy][16 + j0     - lm];
      bY.y = tw[0][dy][16 + j0 + 1 - lm];
      bX.x = tw[1][dy][16 + j0     - lm];
      bX.y = tw[1][dy][16 + j0 + 1 - lm];
      accY = __builtin_amdgcn_wmma_f32_16x16x4_f32(false, a, false, bY, (short)0, accY, false, false);
      accX = __builtin_amdgcn_wmma_f32_16x16x4_f32(false, a, false, bX, (short)0, accX, false, false);
    }
  }

  // epilogue: D layout -> row = r0 + i + 8*lhi, col = c0 + lm
  const float INV_PIO4 = 4.0f / 3.14159265358979323846f;
  #pragma unroll
  for (int i = 0; i < 8; ++i) {
    int r = r0 + i + 8 * lhi;
    int c = c0 + lm;
    float sy = accY[i], sx = accX[i];
    float g  = sqrtf(sx * sx + sy * sy);
    float ph = atan2f(sx, sy + 1e-5f);
    int   q  = (((int)rintf(ph * INV_PIO4)) + 4) & 7;   // round-half-even, mod 8
    int pix  = base + r * WW + c;
    int sb   = NHW + n * 2 * HWN + r * WW + c;
    out[sb]        = sy;       // sobel channel 0 = sobel_y
    out[sb + HWN]  = sx;       // sobel channel 1 = sobel_x
    grad[pix]  = g;
    phase[pix] = (unsigned char)q;
  }
}

// ---------------- NMS + threshold -> b_edges ----------------
__global__ __launch_bounds__(256) void kce_nms(const float* __restrict__ grad,
                                               const unsigned char* __restrict__ phase,
                                               float* __restrict__ out) {
  int i = blockIdx.x * 256 + threadIdx.x;          // exact cover of NHW
  int n = i >> 18;
  int p = i & (HWN - 1);
  int y = p >> 9, x = p & (WW - 1);
  float g = grad[i];
  int ph = phase[i] & 3;
  // neighbor pairs per phase%4: 0:(-1,0)/(1,0) 1:(-1,-1)/(1,1) 2:(0,-1)/(0,1) 3:(-1,1)/(1,-1)
  const int dys[4] = { -1, -1, 0, -1 };
  const int dxs[4] = {  0, -1, -1, 1 };
  int oy = dys[ph], ox = dxs[ph];
  const float* gn = grad + n * HWN;
  int y0 = y + oy, x0 = x + ox, y1 = y - oy, x1 = x - ox;
  float n0 = (y0 >= 0 && y0 < HH && x0 >= 0 && x0 < WW) ? gn[y0 * WW + x0] : 0.0f;  // zero pad
  float n1 = (y1 >= 0 && y1 < HH && x1 >= 0 && x1 < WW) ? gn[y1 * WW + x1] : 0.0f;
  bool mask = (g <= n0) || (g < n1);
  out[i] = (!mask && g > 0.1f) ? 1.0f : 0.0f;      // hysteresis branch is provably dead (LO==HI)
}

extern "C" void kernel_launch(void* const* d_in, const int* in_sizes, int n_in,
                              void* d_out, int out_size, void* d_ws, size_t ws_size,
                              hipStream_t stream) {
  (void)in_sizes; (void)n_in; (void)out_size; (void)ws_size;
  const float* img = (const float*)d_in[0];
  float* out = (float*)d_out;
  unsigned char* ws = (unsigned char*)d_ws;
  unsigned* sc        = (unsigned*)ws;                                  // [0]=min key, [1]=max bits
  float* mag          = (float*)(ws + 64);
  float* grad         = (float*)(ws + 64 + (size_t)NHW * 4);
  unsigned char* phse = (unsigned char*)(ws + 64 + (size_t)NHW * 8);

  kce_init <<<1, 32, 0, stream>>>(sc);
  kce_min  <<<2048, 256, 0, stream>>>(img, sc, 3 * NHW);
  kce_mag  <<<NHW / 256, 256, 0, stream>>>(img, sc, mag);
  kce_sobel<<<16384 / 8, 256, 0, stream>>>(mag, sc, out, grad, phse);
  kce_nms  <<<NHW / 256, 256, 0, stream>>>(grad, phse, out);
}